// SamplingSoftmax_65240553226329
// MI455X (gfx1250) — compile-verified
//
#include <hip/hip_runtime.h>
#include <cstdint>

// ---------------------------------------------------------------------------
// SamplingSoftmax on MI455X (gfx1250)  — B=1024, S=100, C=1000, fp32.
//
// HBM-bound: ~420MB traffic -> ~18us floor @ 23.3 TB/s. No GEMM structure, so
// no WMMA; the CDNA5 features that matter are the async global->LDS stream
// (global_load_async_to_lds_b128, ASYNCcnt), split wait counters, ds_add_f32,
// and wave32 cross-lane reductions.
//
// One block per batch b: 4 waves x 32 lanes.
//   wave w handles samples s = w, w+4, ...      (25 rows each for S=100)
//   lane l handles columns c = 4*l + 128*k, k=0..7 (float4 chunks)
// Per-batch logits (mu, sigma) are preloaded into 64 VGPRs (loop-invariant,
// lane-private columns) so the inner loop's only LDS reads are the noise row.
// Triple-buffered async pipeline: prefetch distance 2, wait ladder
// s_wait_asynccnt 16/8/0. Accumulate p, p^2 in VGPRs; merge across waves with
// ds_add_f32; emit prob = sum(p)/S, var = (sum(p^2) - sum(p)^2/S)/(S-1).
// ---------------------------------------------------------------------------

#define C_DIM   1000
#define C_PAD   1024
#define WAVES   4
#define BLOCK   (WAVES * 32)
#define NBUF    3
#define ROW_BYTES        4096   // padded LDS row buffer
#define ROW_VALID_BYTES  4000   // C_DIM * 4

// dynamic LDS layout (byte offsets)
#define LDS_SUMP   0
#define LDS_SUMP2  4096
#define LDS_ROWBUF 8192
#define LDS_TOTAL  (LDS_ROWBUF + WAVES * NBUF * ROW_BYTES)   // 57344 B

// Issue 8 x global_load_async_to_lds_b128 (512B each across 32 lanes) to copy
// one 4000B row (padded to 4096 in LDS). The 8th chunk's global address is
// clamped inside the row so we never read past the noise buffer; its LDS
// destinations still get (duplicated) in-row data, masked later. Non-temporal:
// the noise stream is touch-once, keep it out of the 192MB L2.
__device__ __forceinline__ void async_prefetch_row(unsigned lds_buf_addr,
                                                   int row_off_bytes,
                                                   unsigned long long gbase,
                                                   int lane) {
  const int      vo = row_off_bytes + lane * 16;
  const unsigned la = lds_buf_addr + (unsigned)(lane * 16);
#pragma unroll
  for (int k = 0; k < 7; ++k) {
    asm volatile("global_load_async_to_lds_b128 %0, %1, %2 th:TH_LOAD_NT"
                 :
                 : "v"(la + (unsigned)(k * 512)), "v"(vo + k * 512), "s"(gbase)
                 : "memory");
  }
  int t = lane * 16 + 3584;
  if (t > ROW_VALID_BYTES - 16) t = ROW_VALID_BYTES - 16;   // clamp inside row
  asm volatile("global_load_async_to_lds_b128 %0, %1, %2 th:TH_LOAD_NT"
               :
               : "v"(la + 3584u), "v"(row_off_bytes + t), "s"(gbase)
               : "memory");
}

__device__ __forceinline__ void wait_async_le16() {
  asm volatile("s_wait_asynccnt 16" ::: "memory");
}
__device__ __forceinline__ void wait_async_le8() {
  asm volatile("s_wait_asynccnt 8" ::: "memory");
}
__device__ __forceinline__ void wait_async_0() {
  asm volatile("s_wait_asynccnt 0" ::: "memory");
}

__global__ void __launch_bounds__(BLOCK)
sampling_softmax_kernel(const float* __restrict__ mean_logits,
                        const float* __restrict__ var_logits,
                        const float* __restrict__ noise,
                        float* __restrict__ out,
                        int B, int S) {
  extern __shared__ char smem[];
  float* s_p  = (float*)(smem + LDS_SUMP);
  float* s_p2 = (float*)(smem + LDS_SUMP2);

  const int b    = blockIdx.x;
  const int tid  = threadIdx.x;
  const int w    = tid >> 5;
  const int lane = tid & 31;

  // absolute LDS byte address of dynamic shared base (flat ptr low 32 bits)
  const unsigned smem_lds = (unsigned)(unsigned long long)(void*)smem;

  // zero the cross-wave accumulators
  for (int c = tid; c < C_PAD; c += BLOCK) {
    s_p[c]  = 0.0f;
    s_p2[c] = 0.0f;
  }

  // preload this lane's mu/sigma columns into registers (loop-invariant).
  // float4 index idx = lane + 32k covers columns 4*idx..4*idx+3; clamp the
  // tail so we never read past row b (values for c>=C_DIM are masked anyway).
  const float4* mrow = (const float4*)(mean_logits + (size_t)b * C_DIM);
  const float4* vrow = (const float4*)(var_logits + (size_t)b * C_DIM);
  float4 mu[8], sg[8];
#pragma unroll
  for (int k = 0; k < 8; ++k) {
    int idx = lane + 32 * k;
    if (idx > (C_DIM / 4) - 1) idx = (C_DIM / 4) - 1;   // clamp to float4 249
    mu[k] = mrow[idx];
    sg[k] = vrow[idx];
  }
  __syncthreads();   // s_p/s_p2 zeroed before any wave's final merge

  const int nrows = (S > w) ? (S - w + WAVES - 1) / WAVES : 0;
  const unsigned long long gbase = (unsigned long long)noise;
  // byte offset of this wave's first row; max ~409.6e6 < 2^31, fits i32
  const long long row0    = ((long long)b * S + w) * (long long)C_DIM * 4;
  const int       rowstep = WAVES * C_DIM * 4;

  const unsigned bufbase = smem_lds + LDS_ROWBUF + (unsigned)(w * NBUF) * ROW_BYTES;

  float4 aP[8], aP2[8];
#pragma unroll
  for (int k = 0; k < 8; ++k) {
    aP[k]  = make_float4(0.f, 0.f, 0.f, 0.f);
    aP2[k] = make_float4(0.f, 0.f, 0.f, 0.f);
  }

  if (nrows > 0) async_prefetch_row(bufbase, (int)row0, gbase, lane);
  if (nrows > 1)
    async_prefetch_row(bufbase + ROW_BYTES, (int)(row0 + rowstep), gbase, lane);

  int bi = 0;   // rotating buffer index of row i
  for (int i = 0; i < nrows; ++i) {
    if (i + 2 < nrows) {
      int bnext = bi + 2; if (bnext >= NBUF) bnext -= NBUF;
      async_prefetch_row(bufbase + (unsigned)bnext * ROW_BYTES,
                         (int)(row0 + (long long)(i + 2) * rowstep),
                         gbase, lane);
      wait_async_le16();   // 2 rows (16 ops) still outstanding -> row i landed
    } else if (i + 1 < nrows) {
      wait_async_le8();    // 1 row outstanding -> row i landed
    } else {
      wait_async_0();
    }

    const float4* nb =
        (const float4*)(smem + LDS_ROWBUF + (size_t)(w * NBUF + bi) * ROW_BYTES);

    // 1) generate samples x = noise*sigma + mu, track row max
    float4 x[8];
    float lmax = -3.0e38f;
#pragma unroll
    for (int k = 0; k < 8; ++k) {
      const int idx = lane + 32 * k;          // float4 index; c0 = 4*idx
      float4 nz = nb[idx];
      float4 xx;
      xx.x = fmaf(nz.x, sg[k].x, mu[k].x);
      xx.y = fmaf(nz.y, sg[k].y, mu[k].y);
      xx.z = fmaf(nz.z, sg[k].z, mu[k].z);
      xx.w = fmaf(nz.w, sg[k].w, mu[k].w);
      if (k == 7) {                            // mask columns >= C_DIM
        const int c0 = 4 * idx;
        if (c0 + 0 >= C_DIM) xx.x = -3.0e38f;
        if (c0 + 1 >= C_DIM) xx.y = -3.0e38f;
        if (c0 + 2 >= C_DIM) xx.z = -3.0e38f;
        if (c0 + 3 >= C_DIM) xx.w = -3.0e38f;
      }
      x[k] = xx;
      lmax = fmaxf(lmax, fmaxf(fmaxf(xx.x, xx.y), fmaxf(xx.z, xx.w)));
    }
#pragma unroll
    for (int off = 16; off > 0; off >>= 1)
      lmax = fmaxf(lmax, __shfl_xor(lmax, off, 32));

    // 2) exponentiate, row sum
    float lsum = 0.0f;
#pragma unroll
    for (int k = 0; k < 8; ++k) {
      x[k].x = __expf(x[k].x - lmax);
      x[k].y = __expf(x[k].y - lmax);
      x[k].z = __expf(x[k].z - lmax);
      x[k].w = __expf(x[k].w - lmax);
      lsum += (x[k].x + x[k].y) + (x[k].z + x[k].w);
    }
#pragma unroll
    for (int off = 16; off > 0; off >>= 1)
      lsum += __shfl_xor(lsum, off, 32);

    // 3) normalize, accumulate p and p^2
    const float inv = __builtin_amdgcn_rcpf(lsum);
#pragma unroll
    for (int k = 0; k < 8; ++k) {
      float4 p;
      p.x = x[k].x * inv;
      p.y = x[k].y * inv;
      p.z = x[k].z * inv;
      p.w = x[k].w * inv;
      aP[k].x += p.x;  aP[k].y += p.y;  aP[k].z += p.z;  aP[k].w += p.w;
      aP2[k].x = fmaf(p.x, p.x, aP2[k].x);
      aP2[k].y = fmaf(p.y, p.y, aP2[k].y);
      aP2[k].z = fmaf(p.z, p.z, aP2[k].z);
      aP2[k].w = fmaf(p.w, p.w, aP2[k].w);
    }

    ++bi; if (bi >= NBUF) bi = 0;
  }

  // merge per-wave accumulators (masked cols contributed exactly 0)
#pragma unroll
  for (int k = 0; k < 8; ++k) {
    const int c0 = 4 * (lane + 32 * k);
    atomicAdd(&s_p[c0 + 0], aP[k].x);
    atomicAdd(&s_p[c0 + 1], aP[k].y);
    atomicAdd(&s_p[c0 + 2], aP[k].z);
    atomicAdd(&s_p[c0 + 3], aP[k].w);
    atomicAdd(&s_p2[c0 + 0], aP2[k].x);
    atomicAdd(&s_p2[c0 + 1], aP2[k].y);
    atomicAdd(&s_p2[c0 + 2], aP2[k].z);
    atomicAdd(&s_p2[c0 + 3], aP2[k].w);
  }
  __syncthreads();

  const float invS   = 1.0f / (float)S;
  const float invSm1 = 1.0f / (float)(S - 1);
  for (int c = tid; c < C_DIM; c += BLOCK) {
    const float sp  = s_p[c];
    const float sp2 = s_p2[c];
    const float prob = sp * invS;
    const float var  = (sp2 - sp * sp * invS) * invSm1;  // unbiased (ddof=1)
    __builtin_nontemporal_store(prob, &out[(size_t)b * C_DIM + c]);
    __builtin_nontemporal_store(var,
        &out[(size_t)B * C_DIM + (size_t)b * C_DIM + c]);
  }
}

extern "C" void kernel_launch(void* const* d_in, const int* in_sizes, int n_in,
                              void* d_out, int out_size, void* d_ws, size_t ws_size,
                              hipStream_t stream) {
  const float* mean_logits = (const float*)d_in[0];
  const float* var_logits  = (const float*)d_in[1];
  const float* noise       = (const float*)d_in[2];
  float*       out         = (float*)d_out;

  const int BC = in_sizes[0];                      // B*C
  const int B  = BC / C_DIM;                       // 1024
  const int S  = (BC > 0) ? in_sizes[2] / BC : 0;  // 100

  hipLaunchKernelGGL(sampling_softmax_kernel, dim3(B), dim3(BLOCK), LDS_TOTAL,
                     stream, mean_logits, var_logits, noise, out, B, S);
}